// SpecAugment_63651415327035
// MI455X (gfx1250) — compile-verified
//
#include <hip/hip_runtime.h>
#include <stdint.h>

// ---------------------------------------------------------------------------
// SpecAugment for x[B=128, F=80, T=3000] f32.
//   out = x with 2 random freq bands (per batch) and 2 random time bands
//   (per batch) zeroed.  Memory-bound: ~246 MB traffic -> ~10.5us @ 23.3TB/s.
// CDNA5 paths: TDM tensor_load_to_lds (double-buffered, 2 rows/block) with
// s_wait_tensorcnt pipelining, ds_load_b128 LDS reads, NT b128 stores.
// Freq-masked rows skip the HBM read entirely (write-only zeros).
// ---------------------------------------------------------------------------

#define B_DIM 128
#define F_DIM 80
#define T_DIM 3000
#define T_VEC (T_DIM / 4)          // 750 float4 per row
#define FREQ_MASK_PARAM 27
#define TIME_MASK_PARAM 100

typedef float f32x4 __attribute__((ext_vector_type(4)));
typedef unsigned int v4u __attribute__((ext_vector_type(4)));
typedef int v4i __attribute__((ext_vector_type(4)));
typedef int v8i __attribute__((ext_vector_type(8)));

// ------------------------- Threefry-2x32 (20 rounds) -----------------------
__device__ __forceinline__ uint32_t rotl32(uint32_t x, int r) {
  return (x << r) | (x >> (32 - r));
}

__device__ __forceinline__ void tf2x32(uint32_t k0, uint32_t k1,
                                       uint32_t x0, uint32_t x1,
                                       uint32_t& o0, uint32_t& o1) {
  const int R[8] = {13, 15, 26, 6, 17, 29, 16, 24};
  uint32_t ks2 = k0 ^ k1 ^ 0x1BD11BDAu;
  x0 += k0; x1 += k1;
#pragma unroll
  for (int i = 0; i < 4; i++) { x0 += x1; x1 = rotl32(x1, R[i]); x1 ^= x0; }
  x0 += k1; x1 += ks2 + 1u;
#pragma unroll
  for (int i = 4; i < 8; i++) { x0 += x1; x1 = rotl32(x1, R[i]); x1 ^= x0; }
  x0 += ks2; x1 += k0 + 2u;
#pragma unroll
  for (int i = 0; i < 4; i++) { x0 += x1; x1 = rotl32(x1, R[i]); x1 ^= x0; }
  x0 += k0; x1 += k1 + 3u;
#pragma unroll
  for (int i = 4; i < 8; i++) { x0 += x1; x1 = rotl32(x1, R[i]); x1 ^= x0; }
  x0 += k1; x1 += ks2 + 4u;
#pragma unroll
  for (int i = 0; i < 4; i++) { x0 += x1; x1 = rotl32(x1, R[i]); x1 ^= x0; }
  x0 += ks2; x1 += k0 + 5u;
  o0 = x0; o1 = x1;
}

// jax.random.split: bits = threefry(key, iota(4)) pairing (0,2),(1,3);
// childA = (o0_l0, o0_l1), childB = (o1_l0, o1_l1)
__device__ __forceinline__ void tf_split(uint32_t k0, uint32_t k1,
                                         uint32_t& a0, uint32_t& a1,
                                         uint32_t& b0, uint32_t& b1) {
  uint32_t x0, x1, y0, y1;
  tf2x32(k0, k1, 0u, 2u, x0, x1);
  tf2x32(k0, k1, 1u, 3u, y0, y1);
  a0 = x0; a1 = y0;
  b0 = x1; b1 = y1;
}

// random_bits for a flat array of 256 counters: lane j%128 processes counters
// (lane, lane+128); flat[lane]=o0, flat[lane+128]=o1.
__device__ __forceinline__ uint32_t rb256(uint32_t k0, uint32_t k1, int j) {
  uint32_t o0, o1;
  uint32_t lane = (uint32_t)(j & 127);
  tf2x32(k0, k1, lane, lane + 128u, o0, o1);
  return (j < 128) ? o0 : o1;
}

// -------------------- mask-parameter generator (1 block) -------------------
// params[b*8 + {0..7}] = fs0, fw0, fs1, fw1, ts0, tw0, ts1, tw1
__global__ void specaug_maskgen(int* __restrict__ params) {
  int b = threadIdx.x;               // 0..127, one batch per thread
  if (b >= B_DIM) return;

  uint32_t kf0, kf1, kt0, kt1;
  tf_split(0u, 42u, kf0, kf1, kt0, kt1);   // jax.random.key(42) -> (kf, kt)

  int* p = params + b * 8;

  {  // frequency masks
    uint32_t kw0, kw1, ko0, ko1;
    tf_split(kf0, kf1, kw0, kw1, ko0, ko1);
#pragma unroll
    for (int m = 0; m < 2; m++) {
      uint32_t wb = rb256(kw0, kw1, 2 * b + m);
      int w = (int)(wb % (uint32_t)(FREQ_MASK_PARAM + 1));
      if (w > F_DIM) w = F_DIM;
      uint32_t ub = rb256(ko0, ko1, 2 * b + m);
      float u = (float)(ub >> 9) * (1.0f / 8388608.0f);   // [0,1)
      int span = F_DIM - w; if (span < 1) span = 1;
      int s = (int)floorf(u * (float)span);
      p[2 * m + 0] = s;
      p[2 * m + 1] = w;
    }
  }
  {  // time masks
    uint32_t kw0, kw1, ko0, ko1;
    tf_split(kt0, kt1, kw0, kw1, ko0, ko1);
#pragma unroll
    for (int m = 0; m < 2; m++) {
      uint32_t wb = rb256(kw0, kw1, 2 * b + m);
      int w = (int)(wb % (uint32_t)(TIME_MASK_PARAM + 1));
      if (w > T_DIM) w = T_DIM;
      uint32_t ub = rb256(ko0, ko1, 2 * b + m);
      float u = (float)(ub >> 9) * (1.0f / 8388608.0f);
      int span = T_DIM - w; if (span < 1) span = 1;
      int s = (int)floorf(u * (float)span);
      p[4 + 2 * m + 0] = s;
      p[4 + 2 * m + 1] = w;
    }
  }
}

// --------------------------- TDM helper (1 row) ----------------------------
#if __has_builtin(__builtin_amdgcn_tensor_load_to_lds)
#define SPECAUG_USE_TDM 1
__device__ __forceinline__ void tdm_row_to_lds(const float* gsrc, uint32_t laddr) {
  const uint64_t ga = (uint64_t)(uintptr_t)gsrc;

  // D# group 0: count=1, lds_addr, global_addr[56:0], type=2 in [127:126]
  v4u g0;
  g0.x = 1u;
  g0.y = laddr;
  g0.z = (uint32_t)ga;
  g0.w = (uint32_t)((ga >> 32) & 0x1FFFFFFu) | (2u << 30);

  // D# group 1: data_size=4B(code 2); tensor_dim0=3000; tensor_dim1=1;
  // tile_dim0=3000; tile_dim1=1; tensor_dim0_stride=3000; no pad/iter/mask.
  v8i g1;
  g1[0] = (int)(2u << 16);                                // data_size
  g1[1] = (int)(((uint32_t)T_DIM & 0xFFFFu) << 16);       // tensor_dim0[15:0]
  g1[2] = (int)(((uint32_t)T_DIM >> 16) | (1u << 16));    // dim0 hi | dim1 lo
  g1[3] = (int)(((uint32_t)T_DIM & 0xFFFFu) << 16);       // dim1 hi | tile_dim0
  g1[4] = 1;                                              // tile_dim1=1
  g1[5] = T_DIM;                                          // dim0_stride lo32
  g1[6] = 0;
  g1[7] = 0;

  const v4i z4 = {0, 0, 0, 0};
#if defined(__clang_major__) && (__clang_major__ >= 23)
  const v8i z8 = {0, 0, 0, 0, 0, 0, 0, 0};
  __builtin_amdgcn_tensor_load_to_lds(g0, g1, z4, z4, z8, 0);
#else
  __builtin_amdgcn_tensor_load_to_lds(g0, g1, z4, z4, 0);
#endif
}
#endif

// ------------------------------- main kernel -------------------------------
// Two consecutive rows (same batch: F=80 even) per workgroup, double-buffered:
// issue both TDMs, s_wait_tensorcnt(1) -> compute row0 while row1 DMA is in
// flight, s_wait_tensorcnt(0) -> compute row1.  Freq-masked rows: write-only.
__global__ void __launch_bounds__(128)
specaug_apply(const float* __restrict__ x, float* __restrict__ out,
              const int* __restrict__ params) {
  const int row0 = blockIdx.x * 2;           // even; row1 = row0+1 same batch
  const int b = row0 / F_DIM;
  const int f0 = row0 - b * F_DIM;
  const int f1 = f0 + 1;

  const int* p = params + b * 8;
  const int fs0 = p[0], fw0 = p[1], fs1 = p[2], fw1 = p[3];
  const int ts0 = p[4], tw0 = p[5], ts1 = p[6], tw1 = p[7];

  const bool live0 = !(((unsigned)(f0 - fs0) < (unsigned)fw0) ||
                       ((unsigned)(f0 - fs1) < (unsigned)fw1));
  const bool live1 = !(((unsigned)(f1 - fs0) < (unsigned)fw0) ||
                       ((unsigned)(f1 - fs1) < (unsigned)fw1));

  const size_t base0 = (size_t)row0 * T_DIM;
  const size_t base1 = base0 + T_DIM;
  f32x4* __restrict__ out0 = (f32x4*)(out + base0);
  f32x4* __restrict__ out1 = (f32x4*)(out + base1);

  __shared__ f32x4 buf[2][T_VEC];            // 24 KB of the 320 KB WGP LDS

#if SPECAUG_USE_TDM
  if (threadIdx.x < 32) {                    // one wave issues the DMAs
    if (live0) tdm_row_to_lds(x + base0, (uint32_t)(uintptr_t)(&buf[0][0]));
    if (live1) tdm_row_to_lds(x + base1, (uint32_t)(uintptr_t)(&buf[1][0]));
  }
#else
  for (int i = threadIdx.x; i < T_VEC; i += blockDim.x) {
    if (live0) buf[0][i] = ((const f32x4*)(x + base0))[i];
    if (live1) buf[1][i] = ((const f32x4*)(x + base1))[i];
  }
#endif

  const f32x4 z = {0.f, 0.f, 0.f, 0.f};

  // ---- row 0 ----
  if (live0) {
#if SPECAUG_USE_TDM
    if (threadIdx.x < 32) {
      if (live1) __builtin_amdgcn_s_wait_tensorcnt(1);   // row0 done, row1 in flight
      else       __builtin_amdgcn_s_wait_tensorcnt(0);
    }
#endif
    __syncthreads();                          // block-uniform path
    for (int i = threadIdx.x; i < T_VEC; i += blockDim.x) {
      f32x4 v = buf[0][i];                    // ds_load_b128
      const int t0 = 4 * i;
#pragma unroll
      for (int c = 0; c < 4; c++) {
        const int tc = t0 + c;
        if (((unsigned)(tc - ts0) < (unsigned)tw0) ||
            ((unsigned)(tc - ts1) < (unsigned)tw1)) v[c] = 0.0f;
      }
      __builtin_nontemporal_store(v, out0 + i);
    }
  } else {
    for (int i = threadIdx.x; i < T_VEC; i += blockDim.x)
      __builtin_nontemporal_store(z, out0 + i);
  }

  // ---- row 1 ----
  if (live1) {
#if SPECAUG_USE_TDM
    if (threadIdx.x < 32) __builtin_amdgcn_s_wait_tensorcnt(0);
#endif
    __syncthreads();                          // block-uniform path
    for (int i = threadIdx.x; i < T_VEC; i += blockDim.x) {
      f32x4 v = buf[1][i];                    // ds_load_b128
      const int t0 = 4 * i;
#pragma unroll
      for (int c = 0; c < 4; c++) {
        const int tc = t0 + c;
        if (((unsigned)(tc - ts0) < (unsigned)tw0) ||
            ((unsigned)(tc - ts1) < (unsigned)tw1)) v[c] = 0.0f;
      }
      __builtin_nontemporal_store(v, out1 + i);
    }
  } else {
    for (int i = threadIdx.x; i < T_VEC; i += blockDim.x)
      __builtin_nontemporal_store(z, out1 + i);
  }
}

// ------------------------------ launch glue --------------------------------
extern "C" void kernel_launch(void* const* d_in, const int* in_sizes, int n_in,
                              void* d_out, int out_size, void* d_ws, size_t ws_size,
                              hipStream_t stream) {
  const float* x = (const float*)d_in[0];     // [128, 80, 3000] f32
  float* out = (float*)d_out;                 // same shape
  int* params = (int*)d_ws;                   // 128 * 8 ints = 4 KB scratch

  specaug_maskgen<<<1, B_DIM, 0, stream>>>(params);
  specaug_apply<<<(B_DIM * F_DIM) / 2, 128, 0, stream>>>(x, out, params);
}